// RegionTokenizer_10282151707092
// MI455X (gfx1250) — compile-verified
//
#include <hip/hip_runtime.h>
#include <math.h>
#include <float.h>

#define BB 8
#define NN 576
#define DD 1024
#define MM 64
#define HM 384
#define WMK 384
#define QD 4096
#define HDIM 1024
#define KR 32
#define PH 24
#define PW 24
#define TOPK 128
#define NVIS 161   // 32 region + 128 context + 1 background

typedef __attribute__((ext_vector_type(2))) float v2f;
typedef __attribute__((ext_vector_type(8))) float v8f;

// ---------------- K1: stable descending argsort of scores (per batch) ----------------
__global__ void k_sort(const float* __restrict__ scores, int* __restrict__ order) {
    int b = blockIdx.x, t = threadIdx.x;
    __shared__ float s[MM];
    s[t] = scores[b * MM + t];
    __syncthreads();
    float v = s[t];
    int rank = 0;
    for (int m = 0; m < MM; m++) {
        float o = s[m];
        rank += (o > v) || (o == v && m < t);
    }
    order[b * MM + rank] = t;
}

// ---------------- K2: 16x16 avg-pool of sorted masks + full-mask sums (bandwidth pass) ----------------
__global__ void k_maskpool(const float* __restrict__ masks, const int* __restrict__ order,
                           float* __restrict__ mg_raw, float* __restrict__ msum) {
    int blk = blockIdx.x;
    int b = blk >> 6, j = blk & 63, t = threadIdx.x;
    int mi = order[b * MM + j];
    const float* mp = masks + ((size_t)(b * MM + mi)) * (HM * WMK);
    float tsum = 0.f;
    for (int c = t; c < PH * PW; c += 256) {
        int y = c / PW, x = c % PW;
        float cs = 0.f;
        for (int i = 0; i < 16; i++) {
            const float4* rp = (const float4*)(mp + (size_t)(y * 16 + i) * WMK + x * 16);
            float4 a = rp[0], b4 = rp[1], c4 = rp[2], d4 = rp[3];
            cs += a.x + a.y + a.z + a.w + b4.x + b4.y + b4.z + b4.w
                + c4.x + c4.y + c4.z + c4.w + d4.x + d4.y + d4.z + d4.w;
        }
        tsum += cs;
        if (j < KR) mg_raw[((b * KR) + j) * (PH * PW) + c] = cs * (1.0f / 256.0f);
    }
    __shared__ float red[256];
    red[t] = tsum;
    __syncthreads();
    for (int s = 128; s > 0; s >>= 1) {
        if (t < s) red[t] += red[t + s];
        __syncthreads();
    }
    if (t == 0) msum[b * MM + mi] = red[0];
}

// ---------------- K3a: grounding_valid + thresholded/fallback scores_k ----------------
__global__ void k_valid(const float* __restrict__ scores, const int* __restrict__ order,
                        const float* __restrict__ msum, float* __restrict__ sk,
                        float* __restrict__ gv, float* __restrict__ out_gv) {
    int b = blockIdx.x, t = threadIdx.x;
    __shared__ int anyf;
    if (t == 0) anyf = 0;
    __syncthreads();
    if (msum[b * MM + t] > 0.f) atomicOr(&anyf, 1);
    __syncthreads();
    int g = anyf;
    if (t == 0) { gv[b] = g ? 1.f : 0.f; out_gv[b] = g ? 1.f : 0.f; }
    if (t < KR) {
        float s = scores[b * MM + order[b * MM + t]];
        float v = (s >= 0.5f) ? s : 0.f;
        if (!g) v = 1.f;
        sk[b * KR + t] = v;
    }
}

// ---------------- K3b: finalize mg (valid/fallback), nm normalization, geometry ----------------
__global__ void k_mgfin(const float* __restrict__ scores, const int* __restrict__ order,
                        const float* __restrict__ gv, float* __restrict__ mg,
                        float* __restrict__ nm, float* __restrict__ geom) {
    int blk = blockIdx.x;
    int b = blk >> 5, j = blk & 31, t = threadIdx.x;
    __shared__ float mgs[NN];
    __shared__ float rsum[256], rcnt[256], ry0[256], ry1[256], rx0[256], rx1[256];
    float g = gv[b];
    float s = scores[b * MM + order[b * MM + j]];
    int valid = s >= 0.5f;
    float psum = 0.f, pcnt = 0.f;
    float y0 = (float)PH, y1 = -1.f, x0 = (float)PW, x1 = -1.f;
    for (int c = t; c < NN; c += 256) {
        float raw = mg[blk * NN + c];
        float v = (g > 0.5f) ? (valid ? raw : 0.f) : (1.0f / NN);
        mgs[c] = v;
        mg[blk * NN + c] = v;
        psum += v;
        if (v > 0.05f) {
            float y = (float)(c / PW), x = (float)(c % PW);
            pcnt += 1.f;
            y0 = fminf(y0, y); y1 = fmaxf(y1, y);
            x0 = fminf(x0, x); x1 = fmaxf(x1, x);
        }
    }
    rsum[t] = psum; rcnt[t] = pcnt;
    ry0[t] = y0; ry1[t] = y1; rx0[t] = x0; rx1[t] = x1;
    __syncthreads();
    for (int st = 128; st > 0; st >>= 1) {
        if (t < st) {
            rsum[t] += rsum[t + st]; rcnt[t] += rcnt[t + st];
            ry0[t] = fminf(ry0[t], ry0[t + st]); ry1[t] = fmaxf(ry1[t], ry1[t + st]);
            rx0[t] = fminf(rx0[t], rx0[t + st]); rx1[t] = fmaxf(rx1[t], rx1[t + st]);
        }
        __syncthreads();
    }
    float tot = rsum[0];
    float inv = 1.0f / fmaxf(tot, 1e-6f);
    for (int c = t; c < NN; c += 256) nm[blk * NN + c] = fmaxf(mgs[c], 0.f) * inv;
    if (t == 0) {
        bool ha = rcnt[0] > 0.f;
        float den = 1.0f / (float)(PW - 1);
        geom[blk * 5 + 0] = ha ? rx0[0] * den : 0.f;
        geom[blk * 5 + 1] = ha ? ry0[0] * den : 0.f;
        geom[blk * 5 + 2] = ha ? rx1[0] * den : 0.f;
        geom[blk * 5 + 3] = ha ? ry1[0] * den : 0.f;
        geom[blk * 5 + 4] = ha ? rcnt[0] * (1.0f / NN) : 0.f;
    }
}

// ---------------- K3c: patch_prior ----------------
__global__ void k_pprior(const float* __restrict__ mg, const float* __restrict__ sk,
                         const float* __restrict__ gv, float* __restrict__ out_pp) {
    int b = blockIdx.x, t = threadIdx.x;       // 576 threads
    int wid = t >> 5, lane = t & 31;
    __shared__ float sks[KR];
    __shared__ float pw[18];
    __shared__ float tot_s;
    if (t < KR) sks[t] = sk[b * KR + t];
    __syncthreads();
    float acc = 0.f;
    for (int j = 0; j < KR; j++) acc += mg[((b * KR) + j) * NN + t] * sks[j];
    acc = fmaxf(acc, 0.f);
    float v = acc;
    for (int m = 16; m > 0; m >>= 1) v += __shfl_xor(v, m);
    if (lane == 0) pw[wid] = v;
    __syncthreads();
    if (t == 0) { float s = 0.f; for (int w = 0; w < 18; w++) s += pw[w]; tot_s = s; }
    __syncthreads();
    float pp = (gv[b] > 0.5f) ? acc / fmaxf(tot_s, 1e-6f) : (1.0f / NN);
    out_pp[b * NN + t] = pp;
}

// ---------------- K5a: h1 = silu(geom @ W1 + b1) ----------------
__global__ void k_h1(const float* __restrict__ geom, const float* __restrict__ W1,
                     const float* __restrict__ b1, float* __restrict__ h1) {
    int bk = blockIdx.x, t = threadIdx.x;
    float g0 = geom[bk * 5 + 0], g1 = geom[bk * 5 + 1], g2 = geom[bk * 5 + 2];
    float g3 = geom[bk * 5 + 3], g4 = geom[bk * 5 + 4];
    for (int i = 0; i < 4; i++) {
        int d = t + 256 * i;
        float x = b1[d] + g0 * W1[d] + g1 * W1[DD + d] + g2 * W1[2 * DD + d]
                + g3 * W1[3 * DD + d] + g4 * W1[4 * DD + d];
        h1[bk * DD + d] = x / (1.0f + expf(-x));
    }
}

// ---------------- K5b: region tokens = sk*(nm@pt) + (h1@W2 + b2) + type_emb[0] ----------------
__global__ void k_region(const float* __restrict__ pt, const float* __restrict__ nm,
                         const float* __restrict__ h1, const float* __restrict__ W2,
                         const float* __restrict__ b2, const float* __restrict__ sk,
                         const float* __restrict__ te, float* __restrict__ out_region,
                         float* __restrict__ out_visual) {
    int dg = blockIdx.x, kg = blockIdx.y, b = blockIdx.z, t = threadIdx.x;
    __shared__ float nms[8 * NN];
    __shared__ float h1s[8 * DD];
    int kbase = kg * 8;
    for (int i = t; i < 8 * NN; i += 256) {
        int kk = i / NN, n = i - kk * NN;
        nms[i] = nm[((b * KR) + kbase + kk) * NN + n];
    }
    for (int i = t; i < 8 * DD; i += 256) {
        int kk = i >> 10, c = i & 1023;
        h1s[i] = h1[((b * KR) + kbase + kk) * DD + c];
    }
    __syncthreads();
    int d = dg * 256 + t;
    float ap[8] = {0}, ag[8] = {0};
    for (int n = 0; n < NN; n++) {
        float p = pt[((size_t)(b * NN) + n) * DD + d];
#pragma unroll
        for (int kk = 0; kk < 8; kk++) ap[kk] += nms[kk * NN + n] * p;
    }
    for (int c = 0; c < DD; c++) {
        float w = W2[(size_t)c * DD + d];
#pragma unroll
        for (int kk = 0; kk < 8; kk++) ag[kk] += h1s[kk * DD + c] * w;
    }
    float bbv = b2[d], t0 = te[d];
    for (int kk = 0; kk < 8; kk++) {
        int k = kbase + kk;
        float val = sk[b * KR + k] * ap[kk] + ag[kk] + bbv + t0;
        out_region[((b * KR) + k) * DD + d] = val;
        out_visual[((size_t)(b * NVIS) + k) * DD + d] = val;
    }
}

// ---------------- K4: q @ Wq partials (deterministic, no atomics) ----------------
__global__ void k_qv(const float* __restrict__ q, const float* __restrict__ Wq,
                     float* __restrict__ qvpart) {
    int dg = blockIdx.x, kg = blockIdx.y, t = threadIdx.x;
    __shared__ float qs[8 * 512];
    for (int i = t; i < 8 * 512; i += 256) {
        int b_ = i >> 9, kk = i & 511;
        qs[i] = q[b_ * QD + kg * 512 + kk];
    }
    __syncthreads();
    int d = dg * 256 + t;
    float acc[8] = {0};
    for (int kk = 0; kk < 512; kk++) {
        float w = Wq[(size_t)(kg * 512 + kk) * HDIM + d];
#pragma unroll
        for (int b_ = 0; b_ < 8; b_++) acc[b_] += qs[b_ * 512 + kk] * w;
    }
    for (int b_ = 0; b_ < 8; b_++) qvpart[((b_ * HDIM) + d) * 8 + kg] = acc[b_];
}

__global__ void k_qv2(const float* __restrict__ part, float* __restrict__ qvo) {
    int i = blockIdx.x * 256 + threadIdx.x;    // < 8192
    float s = 0.f;
    for (int kg = 0; kg < 8; kg++) s += part[i * 8 + kg];
    qvo[i] = s;
}

// ---------------- K6: fused WMMA f32 routing GEMM + tanh + Wo epilogue ----------------
// rs[b,n] partials over 16-col tiles; A staged in LDS (stride 17 to avoid bank conflicts).
__global__ void __launch_bounds__(128)
k_routing_wmma(const float* __restrict__ pt, const float* __restrict__ Wv,
               const float* __restrict__ bv, const float* __restrict__ Wp,
               const float* __restrict__ bp, const float* __restrict__ bq,
               const float* __restrict__ qv, const float* __restrict__ pp,
               const float* __restrict__ Wo, float* __restrict__ rspart) {
    int cg = blockIdx.x;                 // column group (64 cols), 0..15
    int rt = blockIdx.y;                 // row tile (16 rows),   0..35
    int b  = blockIdx.z;
    int t  = threadIdx.x;                // 128 threads = 4 waves
    int wave = t >> 5, lane = t & 31;
    int half = lane >> 4, m = lane & 15;
    int rowbase = rt * 16;
    int col0 = cg * 64 + wave * 16;
    __shared__ float Als[256][17];       // K-chunk x 16 rows, padded
    v8f acc = {0.f, 0.f, 0.f, 0.f, 0.f, 0.f, 0.f, 0.f};
    for (int kc = 0; kc < 4; kc++) {
        int kbase = kc * 256;
        for (int i = 0; i < 32; i++) {
            int idx = i * 128 + t;
            int kk = idx & 255, r = idx >> 8;
            Als[kk][r] = pt[((size_t)(b * NN) + rowbase + r) * DD + kbase + kk];
        }
        __syncthreads();
        const float* Bp = Wv + (size_t)(kbase + 2 * half) * HDIM + col0 + m;
        for (int ks = 0; ks < 64; ks++) {
            int k = ks * 4;
            v2f a, bvec;
            a.x = Als[k + 2 * half][m];
            a.y = Als[k + 1 + 2 * half][m];
            bvec.x = Bp[(size_t)k * HDIM];
            bvec.y = Bp[(size_t)(k + 1) * HDIM];
            acc = __builtin_amdgcn_wmma_f32_16x16x4_f32(false, a, false, bvec,
                                                        (short)0, acc, false, false);
        }
        __syncthreads();
    }
    int d = col0 + m;
    float bias = bv[d] + bp[d] + bq[d] + qv[b * HDIM + d];
    float wpv = Wp[d];
    float wov = Wo[d];
#pragma unroll
    for (int r = 0; r < 8; r++) {
        int row = rowbase + r + 8 * half;
        float x = acc[r] + bias + pp[b * NN + row] * wpv;
        float th = tanhf(x);
        float s = th * wov;
        s += __shfl_xor(s, 1); s += __shfl_xor(s, 2);
        s += __shfl_xor(s, 4); s += __shfl_xor(s, 8);
        if (m == 0) rspart[((size_t)(b * NN) + row) * 64 + (cg * 4 + wave)] = s;
    }
}

__global__ void k_rs2(const float* __restrict__ part, float* __restrict__ rs) {
    int i = blockIdx.x * 256 + threadIdx.x;    // < 4608
    float s = 0.f;
    for (int c = 0; c < 64; c++) s += part[(size_t)i * 64 + c];
    rs[i] = s;
}

// ---------------- K7: softmax + stable top-128 (tie -> lower index) ----------------
__global__ void k_softtop(const float* __restrict__ rs, float* __restrict__ out_rw,
                          float* __restrict__ out_sel, int* __restrict__ seli,
                          float* __restrict__ selmask) {
    int b = blockIdx.x, t = threadIdx.x;   // 576 threads = 18 waves
    int wid = t >> 5, lane = t & 31;
    __shared__ float wv[NN];
    __shared__ float pv[18];
    __shared__ int pi[18];
    __shared__ float m0s, s0s;
    __shared__ int bcast;
    float r = rs[b * NN + t];
    wv[t] = r;
    selmask[b * NN + t] = 0.f;
    float v = r;
    for (int m = 16; m > 0; m >>= 1) v = fmaxf(v, __shfl_xor(v, m));
    if (lane == 0) pv[wid] = v;
    __syncthreads();
    if (t == 0) { float mm = -FLT_MAX; for (int w = 0; w < 18; w++) mm = fmaxf(mm, pv[w]); m0s = mm; }
    __syncthreads();
    float e = expf(r - m0s);
    v = e;
    for (int m = 16; m > 0; m >>= 1) v += __shfl_xor(v, m);
    if (lane == 0) pv[wid] = v;
    __syncthreads();
    if (t == 0) { float ss = 0.f; for (int w = 0; w < 18; w++) ss += pv[w]; s0s = ss; }
    __syncthreads();
    out_rw[b * NN + t] = e / s0s;
    for (int rr = 0; rr < TOPK; rr++) {
        float val = wv[t];
        int idx = t;
        for (int m = 16; m > 0; m >>= 1) {
            float ov = __shfl_xor(val, m);
            int oi = __shfl_xor(idx, m);
            if (ov > val || (ov == val && oi < idx)) { val = ov; idx = oi; }
        }
        if (lane == 0) { pv[wid] = val; pi[wid] = idx; }
        __syncthreads();
        if (t == 0) {
            float bvl = pv[0]; int bix = pi[0];
            for (int w = 1; w < 18; w++)
                if (pv[w] > bvl || (pv[w] == bvl && pi[w] < bix)) { bvl = pv[w]; bix = pi[w]; }
            bcast = bix;
            seli[b * TOPK + rr] = bix;
            out_sel[b * TOPK + rr] = (float)bix;
        }
        __syncthreads();
        if (t == bcast) { wv[t] = -FLT_MAX; selmask[b * NN + t] = 1.f; }
        __syncthreads();
    }
}

// ---------------- K8a: context tokens gather ----------------
__global__ void k_ctx(const float* __restrict__ pt, const int* __restrict__ seli,
                      const float* __restrict__ te, float* __restrict__ out_ctx,
                      float* __restrict__ out_visual) {
    int blk = blockIdx.x;
    int b = blk >> 7, j = blk & 127, t = threadIdx.x;
    int n = seli[b * TOPK + j];
    for (int i = 0; i < 4; i++) {
        int d = t + 256 * i;
        float v = pt[((size_t)(b * NN) + n) * DD + d] + te[DD + d];
        out_ctx[((size_t)(b * TOPK) + j) * DD + d] = v;
        out_visual[((size_t)(b * NVIS) + KR + j) * DD + d] = v;
    }
}

// ---------------- K8b: background token (residual mean of unselected) ----------------
__global__ void k_bg(const float* __restrict__ pt, const float* __restrict__ selmask,
                     const float* __restrict__ te, float* __restrict__ out_bg,
                     float* __restrict__ out_visual) {
    int b = blockIdx.x, t = threadIdx.x;
    float acc[4] = {0.f, 0.f, 0.f, 0.f};
    for (int n = 0; n < NN; n++) {
        float w = 1.0f - selmask[b * NN + n];
        if (w != 0.f) {
#pragma unroll
            for (int i = 0; i < 4; i++) acc[i] += pt[((size_t)(b * NN) + n) * DD + t + 256 * i];
        }
    }
    float inv = 1.0f / 448.0f;   // N - top_k = 576 - 128, exact in f32
    for (int i = 0; i < 4; i++) {
        int d = t + 256 * i;
        float v = acc[i] * inv + te[2 * DD + d];
        out_bg[b * DD + d] = v;
        out_visual[((size_t)(b * NVIS) + 160) * DD + d] = v;
    }
}

extern "C" void kernel_launch(void* const* d_in, const int* in_sizes, int n_in,
                              void* d_out, int out_size, void* d_ws, size_t ws_size,
                              hipStream_t stream) {
    const float* pt     = (const float*)d_in[0];
    const float* masks  = (const float*)d_in[1];
    const float* scores = (const float*)d_in[2];
    const float* q      = (const float*)d_in[3];
    const float* W1     = (const float*)d_in[4];
    const float* b1     = (const float*)d_in[5];
    const float* W2     = (const float*)d_in[6];
    const float* b2     = (const float*)d_in[7];
    const float* Wv     = (const float*)d_in[8];
    const float* bv     = (const float*)d_in[9];
    const float* Wp     = (const float*)d_in[10];
    const float* bp     = (const float*)d_in[11];
    const float* Wq     = (const float*)d_in[12];
    const float* bq     = (const float*)d_in[13];
    const float* Wo     = (const float*)d_in[14];
    const float* te     = (const float*)d_in[16];

    float* out = (float*)d_out;
    const size_t OFF_V   = 0;
    const size_t OFF_R   = 1318912;   // 8*161*1024
    const size_t OFF_C   = 1581056;
    const size_t OFF_BG  = 2629632;
    const size_t OFF_PP  = 2637824;
    const size_t OFF_RW  = 2642432;
    const size_t OFF_SEL = 2647040;
    const size_t OFF_GV  = 2648064;

    float* wsf = (float*)d_ws;
    int*   ws_order   = (int*)(wsf + 0);          // 512
    float* ws_msum    = wsf + 512;                 // 512
    float* ws_sk      = wsf + 1024;                // 256
    float* ws_gv      = wsf + 1280;                // 8
    float* ws_mg      = wsf + 1288;                // 147456
    float* ws_nm      = wsf + 148744;              // 147456
    float* ws_geom    = wsf + 296200;              // 1280
    float* ws_h1      = wsf + 297480;              // 262144
    float* ws_qvpart  = wsf + 559624;              // 65536
    float* ws_qv      = wsf + 625160;              // 8192
    float* ws_rspart  = wsf + 633352;              // 294912
    float* ws_rs      = wsf + 928264;              // 4608
    float* ws_selmask = wsf + 932872;              // 4608
    int*   ws_seli    = (int*)(wsf + 937480);      // 1024

    k_sort<<<BB, MM, 0, stream>>>(scores, ws_order);
    k_maskpool<<<BB * MM, 256, 0, stream>>>(masks, ws_order, ws_mg, ws_msum);
    k_valid<<<BB, MM, 0, stream>>>(scores, ws_order, ws_msum, ws_sk, ws_gv, out + OFF_GV);
    k_mgfin<<<BB * KR, 256, 0, stream>>>(scores, ws_order, ws_gv, ws_mg, ws_nm, ws_geom);
    k_pprior<<<BB, NN, 0, stream>>>(ws_mg, ws_sk, ws_gv, out + OFF_PP);
    k_h1<<<BB * KR, 256, 0, stream>>>(ws_geom, W1, b1, ws_h1);
    k_region<<<dim3(4, 4, BB), 256, 0, stream>>>(pt, ws_nm, ws_h1, W2, b2, ws_sk, te,
                                                 out + OFF_R, out + OFF_V);
    k_qv<<<dim3(4, 8), 256, 0, stream>>>(q, Wq, ws_qvpart);
    k_qv2<<<32, 256, 0, stream>>>(ws_qvpart, ws_qv);
    k_routing_wmma<<<dim3(16, 36, BB), 128, 0, stream>>>(pt, Wv, bv, Wp, bp, bq,
                                                         ws_qv, out + OFF_PP, Wo, ws_rspart);
    k_rs2<<<18, 256, 0, stream>>>(ws_rspart, ws_rs);
    k_softtop<<<BB, NN, 0, stream>>>(ws_rs, out + OFF_RW, out + OFF_SEL, ws_seli, ws_selmask);
    k_ctx<<<BB * TOPK, 256, 0, stream>>>(pt, ws_seli, te, out + OFF_C, out + OFF_V);
    k_bg<<<BB, 256, 0, stream>>>(pt, ws_selmask, te, out + OFF_BG, out + OFF_V);
    (void)in_sizes; (void)n_in; (void)out_size; (void)ws_size;
}